// proposal_layer_32633161515455
// MI455X (gfx1250) — compile-verified
//
#include <hip/hip_runtime.h>
#include <hip/hip_bf16.h>

// ---------------- problem constants ----------------
#define A_     9
#define H_     80
#define W_     120
#define HW_    (H_ * W_)          // 9600
#define B_     32
#define NPER   (A_ * HW_)         // 86400 proposals per image
#define PRE_NMS 100
#define NMS_T   0.3f
#define MIN_SZ  8.0f

#define CHUNK   4096
#define NCHUNK  22                // ceil(86400/4096)
#define CAND    (NCHUNK * 100)    // 2200 candidates per batch
#define TILN    7                 // ceil(100/16) -> 112x112 IoU matrix
#define PADN    (TILN * 16)       // 112

typedef __attribute__((ext_vector_type(2))) float v2f;
typedef __attribute__((ext_vector_type(8))) float v8f;

// 9 base anchors (base 16, ratios .5/1/2, scales 8/16/32) — precomputed from the
// numpy generator (banker's rounding included: hs for ratio .5 is 12).
__constant__ float ANCX1[A_] = {-84.f,-176.f,-360.f,-56.f,-120.f,-248.f,-36.f,-80.f,-168.f};
__constant__ float ANCY1[A_] = {-40.f,-88.f,-184.f,-56.f,-120.f,-248.f,-80.f,-168.f,-344.f};
__constant__ float ANCX2[A_] = { 99.f, 191.f, 375.f, 71.f, 135.f, 263.f, 51.f, 95.f, 183.f};
__constant__ float ANCY2[A_] = { 55.f, 103.f, 199.f, 71.f, 135.f, 263.f, 95.f, 183.f, 359.f};

// Decode proposal i of one batch's delta plane (delta.reshape(B,-1,4) is
// contiguous, so row i is a single float4 load).
__device__ __forceinline__ void decode_one(const float* __restrict__ delta_b, int i,
                                           float imgH, float imgW,
                                           float& x1, float& y1, float& x2, float& y2) {
    int a  = i % A_;
    int s  = i / A_;
    int hh = s / W_;
    int ww = s - hh * W_;
    float sx = (float)(ww * 16);
    float sy = (float)(hh * 16);
    float ax1 = ANCX1[a] + sx, ay1 = ANCY1[a] + sy;
    float ax2 = ANCX2[a] + sx, ay2 = ANCY2[a] + sy;
    float aw = ax2 - ax1 + 1.0f;
    float ah = ay2 - ay1 + 1.0f;
    float acx = ax1 + 0.5f * aw;
    float acy = ay1 + 0.5f * ah;
    const float4 d = ((const float4*)delta_b)[i];
    float pcx = d.x * aw + acx;
    float pcy = d.y * ah + acy;
    float pw  = expf(d.z) * aw;
    float ph  = expf(d.w) * ah;
    x1 = fminf(fmaxf(pcx - 0.5f * pw, 0.0f), imgW - 1.0f);
    x2 = fminf(fmaxf(pcx + 0.5f * pw, 0.0f), imgW - 1.0f);
    y1 = fminf(fmaxf(pcy - 0.5f * ph, 0.0f), imgH - 1.0f);
    y2 = fminf(fmaxf(pcy + 0.5f * ph, 0.0f), imgH - 1.0f);
}

__device__ __forceinline__ bool pair_gt(float va, int ia, float vb, int ib) {
    // "a sorts before b": value desc, index asc on ties (lax.top_k tie-break)
    return (va > vb) || (va == vb && ia < ib);
}

// In-LDS bitonic sort, descending, n must be a power of two.
__device__ __forceinline__ void bitonic_desc(float* sv, int* si, int n, int tid, int nt) {
    for (int k = 2; k <= n; k <<= 1) {
        for (int j = k >> 1; j > 0; j >>= 1) {
            __syncthreads();
            for (int t = tid; t < n; t += nt) {
                int x = t ^ j;
                if (x > t) {
                    float va = sv[t], vb = sv[x];
                    int   ia = si[t], ib = si[x];
                    bool desc_blk = ((t & k) == 0);
                    bool sw = desc_blk ? pair_gt(vb, ib, va, ia)
                                       : pair_gt(va, ia, vb, ib);
                    if (sw) { sv[t] = vb; si[t] = ib; sv[x] = va; si[x] = ia; }
                }
            }
        }
    }
    __syncthreads();
}

// ---------------- stage 1: per-chunk top-100 ----------------
__global__ __launch_bounds__(256) void topk_stage1(const float* __restrict__ score,
                                                   const float* __restrict__ delta,
                                                   const float* __restrict__ img,
                                                   float* __restrict__ cand_val,
                                                   int*   __restrict__ cand_idx) {
    __shared__ float sv[CHUNK];
    __shared__ int   si[CHUNK];
    const int b = blockIdx.y;
    const int c = blockIdx.x;
    const float imgH = img[0], imgW = img[1];
    const float* score_b = score + (size_t)b * 2 * A_ * HW_;
    const float* delta0  = delta;            // keep-mask uses batch 0 only (reference quirk)
    const int base = c * CHUNK;

    for (int t = threadIdx.x; t < CHUNK; t += 256) {
        int g = base + t;
        float v = -INFINITY;
        int  id = 0x7fffffff;                 // pads sort after real -inf entries
        if (g < NPER) {
            id = g;
            // min-size filter from batch 0 decode
            float x1, y1, x2, y2;
            __builtin_prefetch(&((const float4*)delta0)[g], 0, 0);  // global_prefetch_b8
            decode_one(delta0, g, imgH, imgW, x1, y1, x2, y2);
            bool k0 = ((x2 - x1 + 1.0f) >= MIN_SZ) && ((y2 - y1 + 1.0f) >= MIN_SZ);
            if (k0) {
                int a = g % A_;
                int s = g / A_;
                v = score_b[(A_ + a) * HW_ + s];
            }
        }
        sv[t] = v; si[t] = id;
    }
    bitonic_desc(sv, si, CHUNK, threadIdx.x, 256);
    if (threadIdx.x < 100) {
        int o = (b * NCHUNK + c) * 100 + threadIdx.x;
        cand_val[o] = sv[threadIdx.x];
        cand_idx[o] = si[threadIdx.x];
    }
}

// ---------------- stage 2: merge 2200 candidates -> final top-100 ----------------
__global__ __launch_bounds__(256) void topk_stage2(const float* __restrict__ cand_val,
                                                   const int*   __restrict__ cand_idx,
                                                   float* __restrict__ fin_val,
                                                   int*   __restrict__ fin_idx) {
    __shared__ float sv[CHUNK];
    __shared__ int   si[CHUNK];
    const int b = blockIdx.x;
    for (int t = threadIdx.x; t < CHUNK; t += 256) {
        if (t < CAND) { sv[t] = cand_val[b * CAND + t]; si[t] = cand_idx[b * CAND + t]; }
        else          { sv[t] = -INFINITY;              si[t] = 0x7fffffff; }
    }
    bitonic_desc(sv, si, CHUNK, threadIdx.x, 256);
    if (threadIdx.x < 100) {
        fin_val[b * 100 + threadIdx.x] = sv[threadIdx.x];
        fin_idx[b * 100 + threadIdx.x] = si[threadIdx.x];
    }
}

// ---------------- stage 3: NMS (one wave32 per batch, WMMA for area-sum) ----------------
__global__ __launch_bounds__(32) void nms_kernel(const float* __restrict__ delta,
                                                 const float* __restrict__ img,
                                                 const float* __restrict__ fin_val,
                                                 const int*   __restrict__ fin_idx,
                                                 float* __restrict__ out) {
    __shared__ float bx1[PADN], by1[PADN], bx2[PADN], by2[PADN];
    __shared__ float barea[PADN], bval[PADN];
    __shared__ int   keep[PADN];
    __shared__ float iou[PADN][PADN];     // 112*112*4 = 50176 B

    const int b    = blockIdx.x;
    const int lane = threadIdx.x;
    const float imgH = img[0], imgW = img[1];
    const float* delta_b = delta + (size_t)b * 4 * A_ * HW_;

    for (int t = lane; t < PADN; t += 32) {
        float x1 = 0.f, y1 = 0.f, x2 = 0.f, y2 = 0.f, v = -INFINITY;
        if (t < PRE_NMS) {
            v = fin_val[b * PRE_NMS + t];
            int id = fin_idx[b * PRE_NMS + t];
            decode_one(delta_b, id, imgH, imgW, x1, y1, x2, y2);
        }
        bx1[t] = x1; by1[t] = y1; bx2[t] = x2; by2[t] = y2;
        bval[t]  = v;
        barea[t] = (x2 - x1 + 1.0f) * (y2 - y1 + 1.0f);
        keep[t]  = (t < PRE_NMS) && (v > -INFINITY);   // isfinite (values are finite or -inf)
    }
    __syncthreads();

    // den[m][n] = area_m + area_n via V_WMMA_F32_16X16X4_F32:
    //   A(16x4) = [area_i | 1 | 0 | 0],  B(4x16) = [1 ; area_j ; 0 ; 0]
    const v8f czero = {};
    const bool lo = (lane < 16);
    const int  l15 = lane & 15;
    for (int ti = 0; ti < TILN; ++ti) {
        v2f af;
        af.x = lo ? barea[ti * 16 + l15] : 0.0f;   // K=0 (lanes 0-15) / K=2 (lanes 16-31)
        af.y = lo ? 1.0f                : 0.0f;    // K=1 / K=3
        for (int tj = 0; tj < TILN; ++tj) {
            v2f bf;
            bf.x = lo ? 1.0f                : 0.0f; // row K=0 / row K=2
            bf.y = lo ? barea[tj * 16 + l15] : 0.0f; // row K=1 / row K=3
            v8f den = __builtin_amdgcn_wmma_f32_16x16x4_f32(
                false, af, false, bf, (short)0, czero, false, false);
            const int n     = tj * 16 + l15;
            const int mbase = ti * 16 + (lo ? 0 : 8);
            const float nx1 = bx1[n], ny1 = by1[n], nx2 = bx2[n], ny2 = by2[n];
#pragma unroll
            for (int vv = 0; vv < 8; ++vv) {
                int m = mbase + vv;
                float iw = fminf(bx2[m], nx2) - fmaxf(bx1[m], nx1) + 1.0f;
                float ih = fminf(by2[m], ny2) - fmaxf(by1[m], ny1) + 1.0f;
                iw = fmaxf(iw, 0.0f);
                ih = fmaxf(ih, 0.0f);
                float inter = iw * ih;
                iou[m][n] = inter / (den[vv] - inter);
            }
        }
    }
    __syncthreads();

    // Greedy suppression (matches fori_loop body: sup = iou[i]>T & idx>i & keep[i])
    for (int i = 0; i < PRE_NMS; ++i) {
        if (keep[i]) {
            for (int j = i + 1 + lane; j < PRE_NMS; j += 32)
                if (iou[i][j] > NMS_T) keep[j] = 0;
        }
        __syncthreads();
    }

    // Compact kept rows to the front; zeros elsewhere (scatter into zeros buffer).
    float* out_b = out + (size_t)b * PRE_NMS * 5;
    for (int t = lane; t < PRE_NMS * 5; t += 32) out_b[t] = 0.0f;
    __syncthreads();
    if (lane == 0) {
        int p = 0;
        for (int i = 0; i < PRE_NMS; ++i) {
            if (keep[i]) {
                out_b[p * 5 + 0] = bx1[i];
                out_b[p * 5 + 1] = by1[i];
                out_b[p * 5 + 2] = bx2[i];
                out_b[p * 5 + 3] = by2[i];
                out_b[p * 5 + 4] = bval[i];
                ++p;
            }
        }
    }
}

// ---------------- launcher ----------------
extern "C" void kernel_launch(void* const* d_in, const int* in_sizes, int n_in,
                              void* d_out, int out_size, void* d_ws, size_t ws_size,
                              hipStream_t stream) {
    const float* score = (const float*)d_in[0];   // (32, 18, 80, 120)
    const float* delta = (const float*)d_in[1];   // (32, 36, 80, 120)
    const float* img   = (const float*)d_in[2];   // (3,)
    float* out = (float*)d_out;                   // (32, 100, 5)

    // workspace layout (~0.6 MB)
    float* cand_val = (float*)d_ws;                       // B*2200
    int*   cand_idx = (int*)(cand_val + B_ * CAND);       // B*2200
    float* fin_val  = (float*)(cand_idx + B_ * CAND);     // B*100
    int*   fin_idx  = (int*)(fin_val + B_ * 100);         // B*100

    dim3 g1(NCHUNK, B_);
    topk_stage1<<<g1, 256, 0, stream>>>(score, delta, img, cand_val, cand_idx);
    topk_stage2<<<B_, 256, 0, stream>>>(cand_val, cand_idx, fin_val, fin_idx);
    nms_kernel<<<B_, 32, 0, stream>>>(delta, img, fin_val, fin_idx, out);
}